// GINVirtualNode_86423331930333
// MI455X (gfx1250) — compile-verified
//
#include <hip/hip_runtime.h>
#include <stdint.h>

typedef float v2f __attribute__((ext_vector_type(2)));
typedef float v8f __attribute__((ext_vector_type(8)));

#define HDIM 128
#define SLOPE 0.2f
#define BN_INV 0.9999950000374997f   // 1/sqrt(1+1e-5)

__device__ __forceinline__ float leaky_(float v) { return v > 0.0f ? v : SLOPE * v; }

// ---------------------------------------------------------------------------
// Big WMMA GEMM: Out[M,128] = epi( (A1 (+A2)) @ B[128,128] + bias )
//   epi: optional leaky, optional BN affine (g*os_mul, shift).
// Block: 256 threads (8 waves) -> 64 rows x 128 cols. grid = ceil(M/64).
// A (and A2) tiles staged to LDS with gfx1250 async global->LDS b128 copies;
// each wave preloads its 16-col B tile into 64 VGPRs, then 32 fully-unrolled
// k-steps of 4 ds_load_b64 + 4 v_wmma_f32_16x16x4_f32.
// ---------------------------------------------------------------------------
template<bool HAS_A2, bool ACT, bool HAS_BN>
__global__ __launch_bounds__(256) void wmma_gemm_big(
    const float* __restrict__ A1, const float* __restrict__ A2,
    const float* __restrict__ Bm, const float* __restrict__ bias,
    const float* __restrict__ oscale, const float* __restrict__ oshift, float os_mul,
    float* __restrict__ Out, int M)
{
    constexpr int K = 128, N = 128, PAD = 4, LDW = K + PAD;   // 132-float rows: conflict-free
    __shared__ float smA [64 * LDW];
    __shared__ float smA2[HAS_A2 ? 64 * LDW : 4];

    const int wave = threadIdx.x >> 5;
    const int lane = threadIdx.x & 31;
    const int half = lane >> 4;          // 0: lanes 0-15, 1: lanes 16-31
    const int sub  = lane & 15;
    const int m0   = blockIdx.x * 64;
    const int col  = wave * 16 + sub;    // this lane's output column

    // ---- stage A (and A2) 64x128 tile into LDS: async copy, 1 row per issue
    {
        const unsigned ldsA  = (unsigned)(uintptr_t)&smA[0];
        const unsigned ldsA2 = (unsigned)(uintptr_t)&smA2[0];
#pragma unroll
        for (int i = 0; i < 8; ++i) {
            const int r  = wave * 8 + i;
            const int gr = (m0 + r < M) ? (m0 + r) : (M - 1);   // clamp tail block
            const float*   g = A1 + (size_t)gr * K + lane * 4;
            const unsigned l = ldsA + (unsigned)(r * LDW + lane * 4) * 4u;
            asm volatile("global_load_async_to_lds_b128 %0, %1, off"
                         :: "v"(l), "v"(g) : "memory");
            if (HAS_A2) {
                const float*   g2 = A2 + (size_t)gr * K + lane * 4;
                const unsigned l2 = ldsA2 + (unsigned)(r * LDW + lane * 4) * 4u;
                asm volatile("global_load_async_to_lds_b128 %0, %1, off"
                             :: "v"(l2), "v"(g2) : "memory");
            }
        }
    }

    // ---- preload this wave's B column tile (issues while async copy in flight)
    v2f bf[32];
#pragma unroll
    for (int kk = 0; kk < 32; ++kk) {
        const int ka = kk * 4 + 2 * half;
        bf[kk].x = Bm[(size_t)ka * N + col];
        bf[kk].y = Bm[(size_t)(ka + 1) * N + col];
    }

    asm volatile("s_wait_asynccnt 0x0" ::: "memory");
    __syncthreads();

    // ---- 4 M-tiles per wave, K fully unrolled
    v8f acc[4] = {{}, {}, {}, {}};
#pragma unroll
    for (int kk = 0; kk < 32; ++kk) {
        const int ka = kk * 4 + 2 * half;
        v2f a[4];
#pragma unroll
        for (int t = 0; t < 4; ++t) {
            a[t] = *(const v2f*)&smA[(t * 16 + sub) * LDW + ka];
            if (HAS_A2) {
                v2f c = *(const v2f*)&smA2[(t * 16 + sub) * LDW + ka];
                a[t].x += c.x; a[t].y += c.y;
            }
        }
#pragma unroll
        for (int t = 0; t < 4; ++t)
            acc[t] = __builtin_amdgcn_wmma_f32_16x16x4_f32(
                         false, a[t], false, bf[kk], (short)0, acc[t], false, false);
    }

    // ---- epilogue (divergence only here; WMMAs are done)
    const float bcol = bias[col];
    const float osc  = HAS_BN ? oscale[col] * os_mul : 1.0f;
    const float osh  = HAS_BN ? oshift[col]          : 0.0f;
#pragma unroll
    for (int t = 0; t < 4; ++t) {
#pragma unroll
        for (int i = 0; i < 8; ++i) {
            const int r = m0 + t * 16 + i + 8 * half;   // D VGPR i holds rows i / i+8
            float d = acc[t][i] + bcol;
            if (ACT) d = leaky_(d);
            if (HAS_BN) d = d * osc + osh;
            if (r < M) Out[(size_t)r * N + col] = d;
        }
    }
}

// ---------------------------------------------------------------------------
// Small WMMA GEMM (M=512 shapes): Out = epiA(A) @ B + bias (+Cadd), K=128.
//  epiA: optional per-row scale (1/denom), optional per-col affine (final BN).
//  B addressed as Bm[k*ldbk + n*ldbn] (supports implicit transpose for fc_W.T)
// blockDim = 32*(N/16), grid = M/16.
// ---------------------------------------------------------------------------
template<bool HAS_RS, bool HAS_CS, bool ACT, bool HAS_CADD>
__global__ __launch_bounds__(256) void wmma_gemm_small(
    const float* __restrict__ A1, const float* __restrict__ rowscale,
    const float* __restrict__ colscale, const float* __restrict__ colshift, float cs_mul,
    const float* __restrict__ Bm, int ldbk, int ldbn,
    const float* __restrict__ bias, const float* __restrict__ Cadd,
    float* __restrict__ Out, int M, int N)
{
    constexpr int K = 128;
    const int wave = threadIdx.x >> 5;
    const int lane = threadIdx.x & 31;
    const int half = lane >> 4;
    const int sub  = lane & 15;
    const int m0   = blockIdx.x * 16;
    const int row  = m0 + sub;
    const int col  = wave * 16 + sub;

    const float rs   = HAS_RS ? rowscale[row] : 1.0f;
    const size_t arow = (size_t)row * K;
    __builtin_prefetch(A1 + arow + 64, 0, 0);

    v8f acc = {};
#pragma unroll 8
    for (int k0 = 0; k0 < K; k0 += 4) {
        const int ka = k0 + 2 * half;
        v2f a = *(const v2f*)(A1 + arow + ka);
        if (HAS_RS) { a.x *= rs; a.y *= rs; }
        if (HAS_CS) {
            a.x = a.x * (colscale[ka]     * cs_mul) + colshift[ka];
            a.y = a.y * (colscale[ka + 1] * cs_mul) + colshift[ka + 1];
        }
        v2f b;
        b.x = Bm[(size_t)ka       * ldbk + (size_t)col * ldbn];
        b.y = Bm[(size_t)(ka + 1) * ldbk + (size_t)col * ldbn];
        acc = __builtin_amdgcn_wmma_f32_16x16x4_f32(
                  false, a, false, b, (short)0, acc, false, false);
    }

    const float bcol = bias[col];
#pragma unroll
    for (int i = 0; i < 8; ++i) {
        const int r = m0 + i + 8 * half;
        float d = acc[i] + bcol;
        if (ACT) d = leaky_(d);
        const size_t o = (size_t)r * N + col;
        if (HAS_CADD) d += Cadd[o];
        Out[o] = d;
    }
}

// --------------------------- scatter / pointwise ---------------------------
__global__ void zero_f32(float* __restrict__ p, size_t n) {
    size_t i = (size_t)blockIdx.x * blockDim.x + threadIdx.x;
    if (i < n) p[i] = 0.0f;
}

__global__ void count_kernel(const int* __restrict__ batch, float* __restrict__ counts, int n) {
    int i = blockIdx.x * blockDim.x + threadIdx.x;
    if (i < n) atomicAdd(&counts[batch[i]], 1.0f);
}

__global__ void recip_kernel(const float* __restrict__ counts, float* __restrict__ recip, int b) {
    int i = blockIdx.x * blockDim.x + threadIdx.x;
    if (i < b) recip[i] = 1.0f / fmaxf(counts[i], 1.0f);
}

__global__ void vn_init_kernel(const float* __restrict__ emb, float* __restrict__ vn, int total) {
    int i = blockIdx.x * blockDim.x + threadIdx.x;
    if (i < total) vn[i] = emb[i & (HDIM - 1)];
}

// h[i][:] += vn[batch[i]][:]
__global__ void vn_gather_add(const int* __restrict__ batch, const float* __restrict__ vn,
                              float* __restrict__ h, int n) {
    size_t t = (size_t)blockIdx.x * blockDim.x + threadIdx.x;
    size_t node = t >> 5;
    int j = (int)(t & 31) * 4;
    if (node < (size_t)n) {
        const float4 v = *(const float4*)(vn + (size_t)batch[node] * HDIM + j);
        float4* hp = (float4*)(h + node * HDIM + j);
        float4 hv = *hp;
        hv.x += v.x; hv.y += v.y; hv.z += v.z; hv.w += v.w;
        *hp = hv;
    }
}

// agg[dst[e]][:] += h[src[e]][:]   (L2-resident; dominant cost)
__global__ void edge_scatter(const int* __restrict__ src, const int* __restrict__ dst,
                             const float* __restrict__ h, float* __restrict__ agg, int e) {
    size_t t = (size_t)blockIdx.x * blockDim.x + threadIdx.x;
    size_t edge = t >> 5;
    int j = (int)(t & 31) * 4;
    if (edge < (size_t)e) {
        const int s = src[edge], d = dst[edge];
        const float4 v = *(const float4*)(h + (size_t)s * HDIM + j);
        float* a = agg + (size_t)d * HDIM + j;
        atomicAdd(a + 0, v.x); atomicAdd(a + 1, v.y);
        atomicAdd(a + 2, v.z); atomicAdd(a + 3, v.w);
    }
}

// pooled[batch[i]][:] += h[i][:]
__global__ void pool_scatter(const int* __restrict__ batch, const float* __restrict__ h,
                             float* __restrict__ pooled, int n) {
    size_t t = (size_t)blockIdx.x * blockDim.x + threadIdx.x;
    size_t node = t >> 5;
    int j = (int)(t & 31) * 4;
    if (node < (size_t)n) {
        const float4 v = *(const float4*)(h + node * HDIM + j);
        float* p = pooled + (size_t)batch[node] * HDIM + j;
        atomicAdd(p + 0, v.x); atomicAdd(p + 1, v.y);
        atomicAdd(p + 2, v.z); atomicAdd(p + 3, v.w);
    }
}

static inline void launch_zero(hipStream_t s, float* p, size_t n) {
    hipLaunchKernelGGL(zero_f32, dim3((unsigned)((n + 255) / 256)), dim3(256), 0, s, p, n);
}

extern "C" void kernel_launch(void* const* d_in, const int* in_sizes, int n_in,
                              void* d_out, int out_size, void* d_ws, size_t ws_size,
                              hipStream_t stream) {
    const float* x        = (const float*)d_in[0];
    const int*   edge     = (const int*)  d_in[1];
    const int*   batch    = (const int*)  d_in[2];
    const float* node_W   = (const float*)d_in[3];
    const float* node_b   = (const float*)d_in[4];
    const float* conv_W1  = (const float*)d_in[5];
    const float* conv_b1  = (const float*)d_in[6];
    const float* conv_g   = (const float*)d_in[7];
    const float* conv_bt  = (const float*)d_in[8];
    const float* conv_W2  = (const float*)d_in[9];
    const float* conv_b2  = (const float*)d_in[10];
    const float* vn_emb   = (const float*)d_in[11];
    const float* vn_W1    = (const float*)d_in[12];
    const float* vn_b1    = (const float*)d_in[13];
    const float* vn_W2    = (const float*)d_in[14];
    const float* vn_b2    = (const float*)d_in[15];
    const float* bn_g     = (const float*)d_in[16];
    const float* bn_b     = (const float*)d_in[17];
    const float* fc_W     = (const float*)d_in[18];
    const float* fc_b     = (const float*)d_in[19];

    const int N_   = in_sizes[2];          // 50000
    const int E_   = in_sizes[1] / 2;      // 800000
    const int DOUT = in_sizes[19];         // 64
    const int B_   = out_size / DOUT;      // 512
    const int L_   = in_sizes[6] / HDIM;   // 3
    const int HH   = HDIM * HDIM;

    char* ws = (char*)d_ws;
    float* h      = (float*)ws; ws += (size_t)N_ * HDIM * sizeof(float);
    float* agg    = (float*)ws; ws += (size_t)N_ * HDIM * sizeof(float);
    float* z      = (float*)ws; ws += (size_t)N_ * HDIM * sizeof(float);
    float* pooled = (float*)ws; ws += (size_t)B_ * HDIM * sizeof(float);
    float* tmp    = (float*)ws; ws += (size_t)B_ * HDIM * sizeof(float);
    float* vn     = (float*)ws; ws += (size_t)B_ * HDIM * sizeof(float);
    float* counts = (float*)ws; ws += (size_t)B_ * sizeof(float);
    float* recip  = (float*)ws; ws += (size_t)B_ * sizeof(float);
    (void)ws_size; (void)n_in;

    const int* src = edge;
    const int* dst = edge + E_;

    const dim3 blk256(256);
    const dim3 gBig((unsigned)((N_ + 63) / 64));
    const dim3 gSmall((unsigned)(B_ / 16));
    const size_t nT = (size_t)N_ * 32, eT = (size_t)E_ * 32;
    const dim3 gN((unsigned)((nT + 255) / 256)), gE((unsigned)((eT + 255) / 256));

    // counts / 1/denom / vn init
    launch_zero(stream, counts, (size_t)B_);
    hipLaunchKernelGGL(count_kernel, dim3((N_ + 255) / 256), blk256, 0, stream, batch, counts, N_);
    hipLaunchKernelGGL(recip_kernel, dim3((B_ + 255) / 256), blk256, 0, stream, counts, recip, B_);
    hipLaunchKernelGGL(vn_init_kernel, dim3((B_ * HDIM + 255) / 256), blk256, 0, stream,
                       vn_emb, vn, B_ * HDIM);

    // h = x @ node_W + node_b
    hipLaunchKernelGGL((wmma_gemm_big<false, false, false>), gBig, blk256, 0, stream,
                       x, (const float*)nullptr, node_W, node_b,
                       (const float*)nullptr, (const float*)nullptr, 1.0f, h, N_);

    for (int l = 0; l < L_; ++l) {
        // h += vn[batch]
        hipLaunchKernelGGL(vn_gather_add, gN, blk256, 0, stream, batch, vn, h, N_);
        // agg = segment_sum(h[src], dst)
        launch_zero(stream, agg, (size_t)N_ * HDIM);
        hipLaunchKernelGGL(edge_scatter, gE, blk256, 0, stream, src, dst, h, agg, E_);
        // z = bn( leaky( (h+agg) @ W1 + b1 ) )
        hipLaunchKernelGGL((wmma_gemm_big<true, true, true>), gBig, blk256, 0, stream,
                           h, agg, conv_W1 + (size_t)l * HH, conv_b1 + (size_t)l * HDIM,
                           conv_g + (size_t)l * HDIM, conv_bt + (size_t)l * HDIM, BN_INV,
                           z, N_);
        // h = leaky( z @ W2 + b2 )
        hipLaunchKernelGGL((wmma_gemm_big<false, true, false>), gBig, blk256, 0, stream,
                           z, (const float*)nullptr, conv_W2 + (size_t)l * HH,
                           conv_b2 + (size_t)l * HDIM,
                           (const float*)nullptr, (const float*)nullptr, 1.0f, h, N_);
        // pooled = segment_sum(h, batch)
        launch_zero(stream, pooled, (size_t)B_ * HDIM);
        hipLaunchKernelGGL(pool_scatter, gN, blk256, 0, stream, batch, h, pooled, N_);
        // tmp = leaky( (pooled/denom) @ vn_W1 + vn_b1 )
        hipLaunchKernelGGL((wmma_gemm_small<true, false, true, false>), gSmall, blk256, 0, stream,
                           pooled, recip, (const float*)nullptr, (const float*)nullptr, 1.0f,
                           vn_W1 + (size_t)l * HH, HDIM, 1,
                           vn_b1 + (size_t)l * HDIM, (const float*)nullptr,
                           tmp, B_, HDIM);
        // vn = vn + ( tmp @ vn_W2 + vn_b2 )
        hipLaunchKernelGGL((wmma_gemm_small<false, false, false, true>), gSmall, blk256, 0, stream,
                           tmp, (const float*)nullptr, (const float*)nullptr,
                           (const float*)nullptr, 1.0f,
                           vn_W2 + (size_t)l * HH, HDIM, 1,
                           vn_b2 + (size_t)l * HDIM, vn,
                           vn, B_, HDIM);
    }

    // out = bn(segment_sum(h, batch)) @ fc_W.T + fc_b
    launch_zero(stream, pooled, (size_t)B_ * HDIM);
    hipLaunchKernelGGL(pool_scatter, gN, blk256, 0, stream, batch, h, pooled, N_);
    hipLaunchKernelGGL((wmma_gemm_small<false, true, false, false>), gSmall,
                       dim3(32 * (DOUT / 16)), 0, stream,
                       pooled, (const float*)nullptr, bn_g, bn_b, BN_INV,
                       fc_W, 1, HDIM,                // B[k][n] = fc_W[n*128+k]
                       fc_b, (const float*)nullptr,
                       (float*)d_out, B_, DOUT);
}